// LengthRegulator_43241730736779
// MI455X (gfx1250) — compile-verified
//
#include <hip/hip_runtime.h>

// Length regulator for MI455X (gfx1250, wave32).
// x: (B=32, T_src=1024, H=512) f32; durations: (B, T_src) i32.
// Outputs (concatenated in d_out as f32): expanded (B, max_len, H), mel_masks (B, max_len).
// Pure data-movement op (~310 MB traffic -> ~13us at 23.3 TB/s). No FLOPs -> no WMMA;
// instead use CDNA5 async global->LDS staging (ASYNCcnt) + non-temporal streaming stores
// so the 64MB x tensor stays resident in the 192MB L2 for its ~3.5x reuse.

#define B_DIM  32
#define TSRC   1024
#define H_DIM  512

typedef float v4f __attribute__((ext_vector_type(4)));

// ---------------- Kernel A: per-batch inclusive cumsum of durations ----------------
__global__ void lr_cumsum_kernel(const int* __restrict__ dur, int* __restrict__ cum) {
    __shared__ int s[TSRC];
    const int b   = blockIdx.x;
    const int tid = threadIdx.x;           // 0..1023
    s[tid] = dur[b * TSRC + tid];
    __syncthreads();
    #pragma unroll
    for (int off = 1; off < TSRC; off <<= 1) {
        int v   = s[tid];
        int add = (tid >= off) ? s[tid - off] : 0;
        __syncthreads();
        s[tid] = v + add;
        __syncthreads();
    }
    cum[b * TSRC + tid] = s[tid];          // inclusive scan
}

// ---------------- Kernel B: searchsorted gather + masked zero-pad ----------------
// grid = (max_len, B), block = 128 threads (4 x wave32); each block emits one
// 2KB output row (128 lanes x float4).
__global__ void lr_expand_kernel(const float* __restrict__ x,
                                 const int*   __restrict__ cum,
                                 float*       __restrict__ out_x,
                                 float*       __restrict__ out_mask,
                                 int max_len) {
    const int t   = blockIdx.x;
    const int b   = blockIdx.y;
    const int tid = threadIdx.x;           // 0..127

    const int* __restrict__ c = cum + b * TSRC;
    const int total = c[TSRC - 1];

    const size_t out_row = ((size_t)b * max_len + t) * H_DIM;
    v4f* __restrict__ orow = (v4f*)(out_x + out_row);

    if (t >= total) {
        // Padded region: zero output, mask = 1 (True).
        v4f z = (v4f)0.0f;
        __builtin_nontemporal_store(z, orow + tid);
        if (tid == 0) out_mask[(size_t)b * max_len + t] = 1.0f;
        return;
    }

    // searchsorted(c, t, side='right'): first idx with c[idx] > t (uniform across block).
    int lo = 0, hi = TSRC;
    while (lo < hi) {
        int mid = (lo + hi) >> 1;
        if (c[mid] <= t) lo = mid + 1; else hi = mid;
    }
    const int idx = (lo < TSRC) ? lo : (TSRC - 1);

    const float* __restrict__ row = x + ((size_t)b * TSRC + idx) * H_DIM;

    // Speculative prefetch of the next source row (consecutive t blocks will need it).
    if (idx + 1 < TSRC) __builtin_prefetch(row + H_DIM, 0, 0);

    // Stage the 2KB source row into LDS via CDNA5 async global->LDS copy (ASYNCcnt).
    __shared__ __align__(16) float lds[H_DIM];
    // Generic shared pointer: addr[31:0] is the workgroup-relative LDS byte offset.
    unsigned lds_addr = (unsigned)(size_t)(&lds[tid * 4]);
    unsigned long long gaddr = (unsigned long long)(row + tid * 4);
    asm volatile("global_load_async_to_lds_b128 %0, %1, off"
                 :: "v"(lds_addr), "v"(gaddr)
                 : "memory");
    asm volatile("s_wait_asynccnt 0x0" ::: "memory");
    __syncthreads();   // other 3 waves' async copies complete before cross-read

    // Read back from LDS and stream to the output row with non-temporal stores
    // (write-once data: keep it out of L2 so x retains residency).
    v4f v = *((const v4f*)lds + tid);
    __builtin_nontemporal_store(v, orow + tid);

    if (tid == 0) out_mask[(size_t)b * max_len + t] = 0.0f;
}

extern "C" void kernel_launch(void* const* d_in, const int* in_sizes, int n_in,
                              void* d_out, int out_size, void* d_ws, size_t ws_size,
                              hipStream_t stream) {
    const float* x   = (const float*)d_in[0];
    const int*   dur = (const int*)d_in[1];

    int* cum = (int*)d_ws;                 // B*TSRC ints = 128 KB scratch

    // out_size = B*max_len*H + B*max_len  =>  max_len = out_size / (B*(H+1))
    const int max_len = out_size / (B_DIM * (H_DIM + 1));

    float* out_x    = (float*)d_out;
    float* out_mask = out_x + (size_t)B_DIM * max_len * H_DIM;

    lr_cumsum_kernel<<<B_DIM, TSRC, 0, stream>>>(dur, cum);

    dim3 grid(max_len, B_DIM);
    lr_expand_kernel<<<grid, 128, 0, stream>>>(x, cum, out_x, out_mask, max_len);
}